// Encoder_82403242541241
// MI455X (gfx1250) — compile-verified
//
#include <hip/hip_runtime.h>

// ---------------------------------------------------------------------------
// CDNA5 (gfx1250) point-cloud encoder, fp32 WMMA path.
// HBM-bound workload (~1.5-2 GB traffic vs 43.6 GFLOP) -> keep fp32 accuracy,
// fuse every BN+ReLU into the consumer, and use V_WMMA_F32_16X16X4_F32 for
// all GEMMs.
// ---------------------------------------------------------------------------

typedef __attribute__((ext_vector_type(2))) float v2f;
typedef __attribute__((ext_vector_type(8))) float v8f;

namespace {
constexpr int kL    = 32;
constexpr int kV    = kL * kL * kL;   // 32768 voxels
constexpr int kP    = 8;
constexpr int kN    = kV * kP;        // 262144 points
constexpr int kK    = 16;             // knn
constexpr int kVK   = kV * kK;        // 524288 gathered rows
constexpr int kCIN  = 48;
constexpr int kC    = 96;
constexpr float kEPS = 1e-5f;
}

// -------------------------------- zero init --------------------------------
__global__ void zero_ws_kernel(float* __restrict__ p, long n) {
    long i = (long)blockIdx.x * blockDim.x + threadIdx.x;
    long stride = (long)gridDim.x * blockDim.x;
    for (; i < n; i += stride) p[i] = 0.0f;
}

// ------------------------------ BN finalize --------------------------------
// mean = sum/M ; var = sq/M - mean^2 (biased, matches jnp.var)
__global__ void bn_finalize_kernel(const float* __restrict__ sum,
                                   const float* __restrict__ sq,
                                   const float* __restrict__ gamma,
                                   const float* __restrict__ beta,
                                   float* __restrict__ scale,
                                   float* __restrict__ shift,
                                   float invM) {
    int c = threadIdx.x;
    if (c < kC) {
        float m   = sum[c] * invM;
        float var = sq[c] * invM - m * m;
        float inv = rsqrtf(var + kEPS);
        float sc  = gamma[c] * inv;
        scale[c]  = sc;
        shift[c]  = beta[c] - m * sc;
    }
}

// ------------------------- WMMA GEMM (+stats) ------------------------------
// OUT[M x 96] = op(A)[M x KD] @ W[KD x 96] (+ bias), accumulating per-column
// sum / sum^2 of OUT for the following BatchNorm.
//   op(A): optional row gather (knn) and optional BN(scale,shift)+ReLU.
// 192 threads = 6 waves; wave w owns 16-column panel w; 16 row-tiles/block.
template <int KD, bool GATHER, bool BN_A, bool BIAS>
__global__ __launch_bounds__(192)
void gemm_wmma_kernel(const float* __restrict__ A,
                      const int*   __restrict__ gidx,
                      const float* __restrict__ scA,
                      const float* __restrict__ shA,
                      const float* __restrict__ W,
                      const float* __restrict__ bias,
                      float* __restrict__ OUT,
                      float* __restrict__ gsum,
                      float* __restrict__ gsq) {
    constexpr int TPB   = 16;    // 16x16 row tiles per block => 256 rows
    constexpr int PITCH = 100;   // even pitch keeps v2f LDS loads 8B aligned
    __shared__ float As[16 * PITCH];
    __shared__ float sSum[kC];
    __shared__ float sSq[kC];

    const int t    = threadIdx.x;
    const int wave = t >> 5;            // 0..5
    const int lane = t & 31;
    const int half = lane >> 4;         // 0: lanes 0-15, 1: lanes 16-31
    const int l16  = lane & 15;
    const int col  = wave * 16 + l16;   // output column 0..95

    if (t < kC) { sSum[t] = 0.0f; sSq[t] = 0.0f; }

    // B fragments: v_wmma_f32_16x16x4_f32 B layout — lanes 0-15 hold rows
    // {k0, k0+1}, lanes 16-31 hold rows {k0+2, k0+3}.
    v2f bf[KD / 4];
#pragma unroll
    for (int kk = 0; kk < KD / 4; ++kk) {
        int k0 = 4 * kk + 2 * half;
        bf[kk].x = W[(k0 + 0) * kC + col];
        bf[kk].y = W[(k0 + 1) * kC + col];
    }
    const float bcol = BIAS ? bias[col] : 0.0f;

    const int rowBase = blockIdx.x * (TPB * 16);
    for (int tt = 0; tt < TPB; ++tt) {
        const int row0 = rowBase + tt * 16;
        __syncthreads();
        // Cooperative A-tile stage (gather + BN + ReLU fused on load).
#pragma unroll
        for (int q = 0; q < (16 * KD) / 192; ++q) {
            int e = t + q * 192;
            int r = e / KD, c = e % KD;
            int src = GATHER ? gidx[row0 + r] : (row0 + r);
            float a = A[(long)src * KD + c];
            if (BN_A) a = fmaxf(a * scA[c] + shA[c], 0.0f);
            As[r * PITCH + c] = a;
        }
        __syncthreads();

        v8f acc = {0.f, 0.f, 0.f, 0.f, 0.f, 0.f, 0.f, 0.f};
#pragma unroll
        for (int kk = 0; kk < KD / 4; ++kk) {
            const int koff = 4 * kk + 2 * half;   // A: lanes 16-31 take K+2/K+3
            v2f a;
            a.x = As[l16 * PITCH + koff];
            a.y = As[l16 * PITCH + koff + 1];
            acc = __builtin_amdgcn_wmma_f32_16x16x4_f32(
                false, a, false, bf[kk], (short)0, acc, false, false);
        }

        float s1 = 0.0f, s2 = 0.0f;
#pragma unroll
        for (int i = 0; i < 8; ++i) {
            float y = acc[i] + bcol;              // row = row0 + i + 8*half
            OUT[(long)(row0 + i + 8 * half) * kC + col] = y;
            s1 += y;
            s2 += y * y;
        }
        atomicAdd(&sSum[col], s1);
        atomicAdd(&sSq[col], s2);
    }
    __syncthreads();
    if (t < kC) {
        atomicAdd(&gsum[t], sSum[t]);
        atomicAdd(&gsq[t], sSq[t]);
    }
}

// ----------------------- GridPool scatter reduce ---------------------------
// ReLU output >= 0 so segment-max can use atomicMax on the uint bit pattern
// against a zero-initialized buffer.
__global__ void gridpool_scatter_kernel(const float* __restrict__ y0,
                                        const float* __restrict__ sc,
                                        const float* __restrict__ sh,
                                        const int*   __restrict__ cluster,
                                        const float* __restrict__ coord,
                                        unsigned int* __restrict__ fmax,
                                        float* __restrict__ csum,
                                        float* __restrict__ cnt) {
    long idx = (long)blockIdx.x * blockDim.x + threadIdx.x;
    if (idx >= (long)kN * 24) return;
    int n = (int)(idx / 24), j = (int)(idx % 24);
    int cl = cluster[n];
#pragma unroll
    for (int u = 0; u < 4; ++u) {
        int c = j * 4 + u;
        float v = fmaxf(y0[(long)n * kC + c] * sc[c] + sh[c], 0.0f);
        atomicMax(&fmax[(long)cl * kC + c], __float_as_uint(v));
    }
    if (j == 0) {
        atomicAdd(&csum[cl * 3 + 0], coord[n * 3 + 0]);
        atomicAdd(&csum[cl * 3 + 1], coord[n * 3 + 1]);
        atomicAdd(&csum[cl * 3 + 2], coord[n * 3 + 2]);
        atomicAdd(&cnt[cl], 1.0f);
    }
}

__global__ void coord_mean_kernel(const float* __restrict__ csum,
                                  const float* __restrict__ cnt,
                                  float* __restrict__ out) {
    int i = blockIdx.x * blockDim.x + threadIdx.x;
    if (i < kV * 3) out[i] = csum[i] / cnt[i / 3];
}

// ------------------- K=16 max-pool (BN+ReLU fused) + stats -----------------
__global__ __launch_bounds__(96)
void maxpool_stats_kernel(const float* __restrict__ y3,
                          const float* __restrict__ sc,
                          const float* __restrict__ sh,
                          float* __restrict__ h2,
                          float* __restrict__ gsum,
                          float* __restrict__ gsq) {
    const int c  = threadIdx.x;            // channel 0..95
    const int v0 = blockIdx.x * 32;
    const float scc = sc[c], shc = sh[c];
    float s1 = 0.0f, s2 = 0.0f;
    for (int vv = 0; vv < 32; ++vv) {
        int v = v0 + vv;
        const float* p = y3 + (long)v * kK * kC + c;
        float m = 0.0f;                    // ReLU >= 0 -> 0 is the identity
#pragma unroll
        for (int k = 0; k < kK; ++k)
            m = fmaxf(m, fmaxf(p[(long)k * kC] * scc + shc, 0.0f));
        h2[(long)v * kC + c] = m;
        s1 += m;
        s2 += m * m;
    }
    atomicAdd(&gsum[c], s1);
    atomicAdd(&gsq[c], s2);
}

// -------------------------- residual + BN + ReLU ---------------------------
__global__ void residual_kernel(const float* __restrict__ fin,
                                const float* __restrict__ y4,
                                const float* __restrict__ sc,
                                const float* __restrict__ sh,
                                float* __restrict__ fout) {
    long i = (long)blockIdx.x * blockDim.x + threadIdx.x;
    if (i < (long)kV * kC) {
        int c = (int)(i % kC);
        fout[i] = fmaxf(fin[i] + y4[i] * sc[c] + sh[c], 0.0f);
    }
}

// ---------------------------------------------------------------------------
extern "C" void kernel_launch(void* const* d_in, const int* in_sizes, int n_in,
                              void* d_out, int out_size, void* d_ws, size_t ws_size,
                              hipStream_t stream) {
    // Inputs (setup_inputs order)
    const float* coord   = (const float*)d_in[0];
    const float* feat    = (const float*)d_in[1];
    const float* gp_fc_w = (const float*)d_in[2];
    const float* gp_g    = (const float*)d_in[3];
    const float* gp_b    = (const float*)d_in[4];
    const float* fc1_w   = (const float*)d_in[5];
    const float* n1_g    = (const float*)d_in[6];
    const float* n1_b    = (const float*)d_in[7];
    const float* la_w1   = (const float*)d_in[8];
    const float* la_b1   = (const float*)d_in[9];
    const float* la_g1   = (const float*)d_in[10];
    const float* la_bt1  = (const float*)d_in[11];
    const float* la_w2   = (const float*)d_in[12];
    const float* la_b2   = (const float*)d_in[13];
    const float* la_g2   = (const float*)d_in[14];
    const float* la_bt2  = (const float*)d_in[15];
    const float* n2_g    = (const float*)d_in[16];
    const float* n2_b    = (const float*)d_in[17];
    const float* fc3_w   = (const float*)d_in[18];
    const float* n3_g    = (const float*)d_in[19];
    const float* n3_b    = (const float*)d_in[20];
    const int*   cluster = (const int*)d_in[21];
    const int*   knn_idx = (const int*)d_in[22];

    float* out = (float*)d_out;           // [V*3 coord_p][V*96 f]

    // Workspace layout (floats). Stats stages: gp, then per d: n1,la1,la2,n2,n3.
    float* w = (float*)d_ws;
    auto stat = [&](int stage) { return w + (long)stage * 384; };
    // per-stage: sum(+0) sq(+96) scale(+192) shift(+288)
    long off = 11L * 384;
    float* cnt  = w + off; off += kV;
    float* csum = w + off; off += 3L * kV;
    float* fA   = w + off; off += (long)kV * kC;     // zero-init (atomicMax)
    const long zeroN = off;
    float* fB   = w + off; off += (long)kV * kC;
    float* y1   = w + off; off += (long)kV * kC;
    float* h2   = w + off; off += (long)kV * kC;
    float* y4   = w + off; off += (long)kV * kC;
    float* big1 = w + off; off += (long)kVK * kC;    // y0, then y2
    float* big2 = w + off; off += (long)kVK * kC;    // y3
    (void)off; (void)in_sizes; (void)n_in; (void)out_size; (void)ws_size;

    // 0) zero accumulators + f-max buffer
    zero_ws_kernel<<<4096, 256, 0, stream>>>(w, zeroN);

    // 1) GridPool GEMM: y0 = feat @ gp_fc_w   [N x 96], stats -> stage 0
    gemm_wmma_kernel<48, false, false, false><<<kN / 256, 192, 0, stream>>>(
        feat, nullptr, nullptr, nullptr, gp_fc_w, nullptr,
        big1, stat(0) + 0, stat(0) + 96);
    bn_finalize_kernel<<<1, 128, 0, stream>>>(
        stat(0) + 0, stat(0) + 96, gp_g, gp_b, stat(0) + 192, stat(0) + 288,
        1.0f / (float)kN);

    // 2) scatter: f = segment_max(relu(bn(y0))), coord sums
    gridpool_scatter_kernel<<<(int)(((long)kN * 24 + 255) / 256), 256, 0, stream>>>(
        big1, stat(0) + 192, stat(0) + 288, cluster, coord,
        (unsigned int*)fA, csum, cnt);
    coord_mean_kernel<<<(kV * 3 + 255) / 256, 256, 0, stream>>>(csum, cnt, out);

    // 3) D=2 residual blocks
    float* fcur = fA;
    for (int d = 0; d < 2; ++d) {
        const int sN1 = 1 + 5 * d, sL1 = 2 + 5 * d, sL2 = 3 + 5 * d,
                  sN2 = 4 + 5 * d, sN3 = 5 + 5 * d;
        const long dCC = (long)d * kC * kC;
        const long dC1 = (long)d * kC;

        // y1 = f @ fc1_w[d], stats n1
        gemm_wmma_kernel<96, false, false, false><<<kV / 256, 192, 0, stream>>>(
            fcur, nullptr, nullptr, nullptr, fc1_w + dCC, nullptr,
            y1, stat(sN1) + 0, stat(sN1) + 96);
        bn_finalize_kernel<<<1, 128, 0, stream>>>(
            stat(sN1) + 0, stat(sN1) + 96, n1_g + dC1, n1_b + dC1,
            stat(sN1) + 192, stat(sN1) + 288, 1.0f / (float)kV);

        // y2 = gather(relu(bn_n1(y1)))[knn] @ la_w1[d] + la_b1[d], stats la1
        gemm_wmma_kernel<96, true, true, true><<<kVK / 256, 192, 0, stream>>>(
            y1, knn_idx, stat(sN1) + 192, stat(sN1) + 288, la_w1 + dCC,
            la_b1 + dC1, big1, stat(sL1) + 0, stat(sL1) + 96);
        bn_finalize_kernel<<<1, 128, 0, stream>>>(
            stat(sL1) + 0, stat(sL1) + 96, la_g1 + dC1, la_bt1 + dC1,
            stat(sL1) + 192, stat(sL1) + 288, 1.0f / (float)kVK);

        // y3 = relu(bn_la1(y2)) @ la_w2[d] + la_b2[d], stats la2
        gemm_wmma_kernel<96, false, true, true><<<kVK / 256, 192, 0, stream>>>(
            big1, nullptr, stat(sL1) + 192, stat(sL1) + 288, la_w2 + dCC,
            la_b2 + dC1, big2, stat(sL2) + 0, stat(sL2) + 96);
        bn_finalize_kernel<<<1, 128, 0, stream>>>(
            stat(sL2) + 0, stat(sL2) + 96, la_g2 + dC1, la_bt2 + dC1,
            stat(sL2) + 192, stat(sL2) + 288, 1.0f / (float)kVK);

        // h2 = max_k relu(bn_la2(y3)), stats n2
        maxpool_stats_kernel<<<kV / 32, 96, 0, stream>>>(
            big2, stat(sL2) + 192, stat(sL2) + 288, h2,
            stat(sN2) + 0, stat(sN2) + 96);
        bn_finalize_kernel<<<1, 128, 0, stream>>>(
            stat(sN2) + 0, stat(sN2) + 96, n2_g + dC1, n2_b + dC1,
            stat(sN2) + 192, stat(sN2) + 288, 1.0f / (float)kV);

        // y4 = relu(bn_n2(h2)) @ fc3_w[d], stats n3
        gemm_wmma_kernel<96, false, true, false><<<kV / 256, 192, 0, stream>>>(
            h2, nullptr, stat(sN2) + 192, stat(sN2) + 288, fc3_w + dCC, nullptr,
            y4, stat(sN3) + 0, stat(sN3) + 96);
        bn_finalize_kernel<<<1, 128, 0, stream>>>(
            stat(sN3) + 0, stat(sN3) + 96, n3_g + dC1, n3_b + dC1,
            stat(sN3) + 192, stat(sN3) + 288, 1.0f / (float)kV);

        // f = relu(f + bn_n3(y4)); last depth writes straight into d_out
        float* fnext = (d == 1) ? (out + 3L * kV) : fB;
        residual_kernel<<<((kV * kC) + 255) / 256, 256, 0, stream>>>(
            fcur, y4, stat(sN3) + 192, stat(sN3) + 288, fnext);
        fcur = fnext;
    }
}